// RandomVectorQuantizer_38001870635049
// MI455X (gfx1250) — compile-verified
//
#include <hip/hip_runtime.h>
#include <math.h>

typedef __attribute__((ext_vector_type(2))) float v2f;
typedef __attribute__((ext_vector_type(8))) float v8f;

#define B_   4
#define T_   2048
#define IN_  512
#define G_   8
#define V_   4096
#define D_   16
#define GD_  128          // G*D
#define BT_  8192         // B*T

// ---------------------------------------------------------------------------
// Kernel 1: projection  x = inputs @ W^T   (BT x IN) x (IN x GD) -> (BT x GD)
// One wave32 per 16x16 output tile, K-loop of 512/4 f32 WMMAs.
// ---------------------------------------------------------------------------
__global__ __launch_bounds__(32)
void proj_wmma(const float* __restrict__ inp, const float* __restrict__ W,
               float* __restrict__ x) {
  const int ntn   = GD_ / 16;                 // 8 column tiles
  const int tile  = blockIdx.x;
  const int mbase = (tile / ntn) * 16;
  const int nbase = (tile % ntn) * 16;
  const int lane  = threadIdx.x;
  const int half  = lane >> 4;                // 0: K{0,1}, 1: K{2,3}
  const int l16   = lane & 15;

  // A row for this lane (token), B column for this lane (W row = output col)
  const float* arow = inp + (size_t)(mbase + l16) * IN_ + 2 * half;
  const float* brow = W   + (size_t)(nbase + l16) * IN_ + 2 * half;

  v8f c = {};
  #pragma unroll 4
  for (int k = 0; k < IN_; k += 4) {
    v2f a = *(const v2f*)(arow + k);          // A[M=l16][k+2h, k+2h+1]
    v2f b = *(const v2f*)(brow + k);          // B[k+2h..][N=l16] == W[N][k+2h..]
    c = __builtin_amdgcn_wmma_f32_16x16x4_f32(false, a, false, b,
                                              (short)0, c, false, false);
  }
  // C layout: half 0 -> rows 0..7, half 1 -> rows 8..15, col = nbase+l16
  #pragma unroll
  for (int r = 0; r < 8; ++r)
    x[(size_t)(mbase + 8 * half + r) * GD_ + nbase + l16] = c[r];
}

// ---------------------------------------------------------------------------
// Kernel 2: in-place L2 normalization over D=16 per (token, group)
// ---------------------------------------------------------------------------
__global__ void l2norm_k(float* __restrict__ x) {
  const int i = blockIdx.x * blockDim.x + threadIdx.x;   // bt*G + g
  float* p = x + (size_t)i * D_;
  float s = 0.f;
  #pragma unroll
  for (int d = 0; d < D_; ++d) s += p[d] * p[d];
  const float inv = 1.0f / fmaxf(sqrtf(s), 1e-12f);
  #pragma unroll
  for (int d = 0; d < D_; ++d) p[d] *= inv;
}

// ---------------------------------------------------------------------------
// Kernel 3: scores = xn . codebook^T per group, fused argmax over V=4096.
// One wave32 per (64 tokens, 1 group): B fragments reused over 4 token tiles.
// ---------------------------------------------------------------------------
__global__ __launch_bounds__(32)
void scores_argmax(const float* __restrict__ xn, const float* __restrict__ cb,
                   int* __restrict__ idsbuf) {
  const int wid  = blockIdx.x;
  const int g    = wid % G_;
  const int t0   = (wid / G_) * 64;
  const int lane = threadIdx.x;
  const int half = lane >> 4;
  const int l16  = lane & 15;

  // A fragments: 4 token tiles x 4 k-steps (K = 16 total)
  v2f afrag[4][4];
  #pragma unroll
  for (int mt = 0; mt < 4; ++mt) {
    const float* xr = xn + (size_t)(t0 + mt * 16 + l16) * GD_ + g * D_ + 2 * half;
    #pragma unroll
    for (int d = 0; d < 4; ++d) afrag[mt][d] = *(const v2f*)(xr + 4 * d);
  }

  float best[4][8];
  int   bidx[4][8];
  #pragma unroll
  for (int mt = 0; mt < 4; ++mt)
    #pragma unroll
    for (int r = 0; r < 8; ++r) { best[mt][r] = -3.4e38f; bidx[mt][r] = 0; }

  const float* cbg = cb + (size_t)g * V_ * D_;

  for (int vb = 0; vb < V_; vb += 16) {
    if (vb + 16 < V_)                                    // stream next tile
      __builtin_prefetch(cbg + (size_t)(vb + 16 + l16) * D_, 0, 0);

    const float* brow = cbg + (size_t)(vb + l16) * D_ + 2 * half;
    v2f bf[4];
    #pragma unroll
    for (int d = 0; d < 4; ++d) bf[d] = *(const v2f*)(brow + 4 * d);

    const int colv = vb + l16;                           // this lane's column
    #pragma unroll
    for (int mt = 0; mt < 4; ++mt) {
      v8f c = {};
      #pragma unroll
      for (int d = 0; d < 4; ++d)
        c = __builtin_amdgcn_wmma_f32_16x16x4_f32(false, afrag[mt][d], false,
                                                  bf[d], (short)0, c,
                                                  false, false);
      #pragma unroll
      for (int r = 0; r < 8; ++r) {
        const bool t = c[r] > best[mt][r];               // strict >: keeps lowest idx
        best[mt][r] = t ? c[r] : best[mt][r];
        bidx[mt][r] = t ? colv : bidx[mt][r];
      }
    }
  }

  // Butterfly reduce within each 16-lane half (xor 1,2,4,8 never crosses bit 4)
  #pragma unroll
  for (int mt = 0; mt < 4; ++mt) {
    #pragma unroll
    for (int r = 0; r < 8; ++r) {
      float m  = best[mt][r];
      int   id = bidx[mt][r];
      #pragma unroll
      for (int off = 1; off < 16; off <<= 1) {
        const float om  = __shfl_xor(m, off, 32);
        const int   oid = __shfl_xor(id, off, 32);
        const bool  t   = (om > m) || (om == m && oid < id);  // JAX tie-break
        m  = t ? om  : m;
        id = t ? oid : id;
      }
      best[mt][r] = m; bidx[mt][r] = id;
    }
  }

  if (l16 == 0) {                  // lane 0 -> rows 0..7, lane 16 -> rows 8..15
    #pragma unroll
    for (int mt = 0; mt < 4; ++mt)
      #pragma unroll
      for (int r = 0; r < 8; ++r) {
        const int t = t0 + mt * 16 + 8 * half + r;
        idsbuf[(size_t)t * G_ + g] = bidx[mt][r];
      }
  }
}

// ---------------------------------------------------------------------------
// Kernel 4: gather quantized rows + apply padding mask, emit final ids
// ---------------------------------------------------------------------------
__global__ void gather_k(const int* __restrict__ idsbuf,
                         const int* __restrict__ pad,
                         const float* __restrict__ cb,
                         int* __restrict__ ids_out,
                         float* __restrict__ q_out) {
  const int i = blockIdx.x * blockDim.x + threadIdx.x;   // bt*G + g
  if (i >= BT_ * G_) return;
  const int bt = i / G_;
  const int g  = i % G_;
  const int id = idsbuf[i];
  const int p  = pad[bt];

  ids_out[i] = (p > 0) ? -1 : id;

  float4* dst = (float4*)(q_out + (size_t)i * D_);
  if (p > 0) {
    const float4 z = {0.f, 0.f, 0.f, 0.f};
    dst[0] = z; dst[1] = z; dst[2] = z; dst[3] = z;
  } else {
    const float4* s4 = (const float4*)(cb + ((size_t)g * V_ + id) * D_);
    dst[0] = s4[0]; dst[1] = s4[1]; dst[2] = s4[2]; dst[3] = s4[3];
  }
}

// ---------------------------------------------------------------------------
extern "C" void kernel_launch(void* const* d_in, const int* in_sizes, int n_in,
                              void* d_out, int out_size, void* d_ws, size_t ws_size,
                              hipStream_t stream) {
  const float* inp = (const float*)d_in[0];   // (B,T,IN) f32
  const int*   pad = (const int*)  d_in[1];   // (B,T) i32
  const float* W   = (const float*)d_in[2];   // (GD,IN) f32
  const float* cb  = (const float*)d_in[3];   // (G,V,D) f32

  float* xn     = (float*)d_ws;                                        // 4 MB
  int*   idsbuf = (int*)((char*)d_ws + (size_t)BT_ * GD_ * sizeof(float));

  int*   ids_out = (int*)d_out;                       // B*T*G int32
  float* q_out   = (float*)d_out + (size_t)BT_ * G_;  // B*T*G*D f32

  proj_wmma    <<<(BT_ / 16) * (GD_ / 16), 32, 0, stream>>>(inp, W, xn);
  l2norm_k     <<<(BT_ * G_) / 256, 256, 0, stream>>>(xn);
  scores_argmax<<<(BT_ / 64) * G_, 32, 0, stream>>>(xn, cb, idsbuf);
  gather_k     <<<(BT_ * G_) / 256, 256, 0, stream>>>(idsbuf, pad, cb,
                                                      ids_out, q_out);
}